// DirModel_39599598469810
// MI455X (gfx1250) — compile-verified
//
#include <hip/hip_runtime.h>

// ---------------------------------------------------------------------------
// DirModel on MI455X (gfx1250): fp32 WMMA 16x16x4, LDS-staged B operand,
// NT-hinted streaming of the 268MB Di/DiA operators, branchless ELU.
// B=2, N=1024, Fc=2048, C=64, 5 blocks.  All big matmuls are [M,K]@[K,16].
// ---------------------------------------------------------------------------

typedef float v2f __attribute__((ext_vector_type(2)));
typedef float v8f __attribute__((ext_vector_type(8)));

// Branchless ELU: v_exp_f32 + v_cndmask, no EXEC-masked branches around WMMA.
__device__ __forceinline__ float eluf(float x) {
    float e = __expf(x) - 1.f;
    return x > 0.f ? x : e;
}

__device__ __forceinline__ v8f wmma4(v2f a, v2f b, v8f c) {
    // V_WMMA_F32_16X16X4_F32 : D[16x16] = A[16x4] * B[4x16] + C
    return __builtin_amdgcn_wmma_f32_16x16x4_f32(
        /*neg_a=*/false, a, /*neg_b=*/false, b,
        /*c_mod=*/(short)0, c, /*reuse_a=*/false, /*reuse_b=*/false);
}

// ---------------------------------------------------------------------------
// v[row,c] = inputs[row,0:3] @ W_in + b_in     (rows = B*N = 2048, c = 64)
// ---------------------------------------------------------------------------
__global__ void projIn(const float* __restrict__ x, const float* __restrict__ W,
                       const float* __restrict__ b, float* __restrict__ v) {
    int i   = blockIdx.x * 256 + threadIdx.x;   // 2048*64 threads
    int row = i >> 6;
    int c   = i & 63;
    v[i] = x[row * 3 + 0] * W[c] + x[row * 3 + 1] * W[64 + c] +
           x[row * 3 + 2] * W[128 + c] + b[c];
}

__global__ void zeroK(float* __restrict__ p, int n) {
    int i = blockIdx.x * 256 + threadIdx.x;
    if (i < n) p[i] = 0.f;
}

// ---------------------------------------------------------------------------
// BatchNorm training-mode stats of elu(in) over R rows, per channel (64).
// Emits folded affine: stats[c] = a = g*rsqrt(var+eps), stats[64+c] = be - m*a
// ---------------------------------------------------------------------------
__global__ void bnStats(const float* __restrict__ in, int R,
                        const float* __restrict__ g, const float* __restrict__ be,
                        float* __restrict__ stats) {
    __shared__ float s1[256], s2[256];
    int c = blockIdx.x, tid = threadIdx.x;
    float sum = 0.f, sq = 0.f;
    for (int r = tid; r < R; r += 256) {
        float e = eluf(in[(size_t)r * 64 + c]);
        sum += e;
        sq  += e * e;
    }
    s1[tid] = sum; s2[tid] = sq;
    __syncthreads();
    for (int st = 128; st > 0; st >>= 1) {
        if (tid < st) { s1[tid] += s1[tid + st]; s2[tid] += s2[tid + st]; }
        __syncthreads();
    }
    if (tid == 0) {
        float m   = s1[0] / (float)R;
        float var = s2[0] / (float)R - m * m;   // biased variance
        float a   = g[c] * rsqrtf(var + 1e-5f);
        stats[c]      = a;
        stats[64 + c] = be[c] - m * a;
    }
}

// ---------------------------------------------------------------------------
// out[R,64] = elu( (elu(in)*a + s) @ W[64,64] + bias )      via WMMA f32
// grid.x = R/32, block = 256 (8 waves): wave -> (2 mtiles per block, 4 ntiles)
// ---------------------------------------------------------------------------
__global__ void transformK(const float* __restrict__ in,
                           const float* __restrict__ stats,
                           const float* __restrict__ W,
                           const float* __restrict__ bias,
                           float* __restrict__ out) {
    __shared__ float S[128];                    // a[64] | s[64]
    int tid = threadIdx.x;
    if (tid < 128) S[tid] = stats[tid];
    __syncthreads();

    int wave  = tid >> 5, lane = tid & 31;
    int lhalf = lane >> 4, lmod = lane & 15;
    int m0 = (blockIdx.x * 2 + (wave >> 2)) * 16;
    int n0 = (wave & 3) * 16;

    const float* irow = in + (size_t)(m0 + lmod) * 64;

    v8f acc;
#pragma unroll
    for (int j = 0; j < 8; j++) acc[j] = 0.f;

#pragma unroll
    for (int k = 0; k < 64; k += 4) {
        int kA = k + 2 * lhalf;                 // A fragment: (M=lmod, K=kA,kA+1)
        v2f iv = *(const v2f*)(irow + kA);      // 8B aligned (kA even)
        v2f a;
        a[0] = eluf(iv[0]) * S[kA]     + S[64 + kA];
        a[1] = eluf(iv[1]) * S[kA + 1] + S[64 + kA + 1];
        v2f b;                                  // B fragment: (K=kA,kA+1, N=n0+lmod)
        b[0] = W[(size_t)kA * 64 + n0 + lmod];
        b[1] = W[(size_t)(kA + 1) * 64 + n0 + lmod];
        acc = wmma4(a, b, acc);
    }

    float bb = bias[n0 + lmod];
    float* orow = out + (size_t)m0 * 64 + n0 + lmod;
#pragma unroll
    for (int j = 0; j < 8; j++)                 // C layout: VGPRj -> M = j + 8*lhalf
        orow[(size_t)(j + 8 * lhalf) * 64] = eluf(acc[j] + bb);
}

// ---------------------------------------------------------------------------
// C[M,16] += A[M,K] @ B[K,16]   (batched via grid.y; accumulator preloaded)
// 8 waves per WG, each owns one 16-row M tile; B staged through LDS in 16 KB
// chunks (K-chunk = 256) and reused by all waves.  A is a one-shot 23TB/s
// stream -> non-temporal loads keep B / f / v resident in the 192MB L2.
// ---------------------------------------------------------------------------
#define KC 256
__global__ void gemmN16(const float* __restrict__ A, const float* __restrict__ Bm,
                        float* __restrict__ C, int M, int K) {
    __shared__ float Bs[KC * 16];               // 16 KB

    int bz = blockIdx.y;
    A  += (size_t)bz * (size_t)M * (size_t)K;
    Bm += (size_t)bz * (size_t)K * 16;
    C  += (size_t)bz * (size_t)M * 16;

    int tid   = threadIdx.x;
    int wave  = tid >> 5, lane = tid & 31;
    int lhalf = lane >> 4, lmod = lane & 15;
    int m0 = (blockIdx.x * 8 + wave) * 16;

    const float* Arow = A + (size_t)(m0 + lmod) * (size_t)K;
    float*       Crow = C + (size_t)m0 * 16;

    v8f acc;                                    // preload accumulator (C +=)
#pragma unroll
    for (int j = 0; j < 8; j++)
        acc[j] = Crow[(size_t)(j + 8 * lhalf) * 16 + lmod];

    for (int k0 = 0; k0 < K; k0 += KC) {
        __syncthreads();                        // previous chunk fully consumed
        {   // cooperative B chunk load: thread t -> row (k0 + t), 16 floats
            const float4* src = (const float4*)(Bm + (size_t)(k0 + tid) * 16);
            float4*       dst = (float4*)(Bs + tid * 16);
            dst[0] = src[0]; dst[1] = src[1]; dst[2] = src[2]; dst[3] = src[3];
        }
        __syncthreads();

#pragma unroll 4
        for (int kk = 0; kk < KC; kk += 4) {
            int kA = k0 + kk + 2 * lhalf;
            // streamed A fragment: non-temporal global_load_b64, 8B aligned
            v2f a = __builtin_nontemporal_load((const v2f*)(Arow + kA));
            int bi = (kk + 2 * lhalf) * 16 + lmod;
            v2f b;
            b[0] = Bs[bi];
            b[1] = Bs[bi + 16];
            acc = wmma4(a, b, acc);
        }
    }

#pragma unroll
    for (int j = 0; j < 8; j++)
        Crow[(size_t)(j + 8 * lhalf) * 16 + lmod] = acc[j];
}

// ---------------------------------------------------------------------------
// Masked mean pool over N=1024, FC 64->10, log_softmax.  1 block / batch.
// ---------------------------------------------------------------------------
__global__ void headK(const float* __restrict__ v2, const float* __restrict__ mask,
                      const float* __restrict__ Wfc, const float* __restrict__ bfc,
                      float* __restrict__ out) {
    __shared__ float pooled[64];
    __shared__ float lg[10];
    __shared__ float msum_s;
    int b = blockIdx.x, tid = threadIdx.x;      // 64 threads
    const float* vb = v2 + (size_t)b * 1024 * 64;
    const float* mb = mask + (size_t)b * 1024;

    float s = 0.f;
    for (int n = 0; n < 1024; n++) s += vb[(size_t)n * 64 + tid] * mb[n];
    if (tid == 0) {
        float ms = 0.f;
        for (int n = 0; n < 1024; n++) ms += mb[n];
        msum_s = ms;
    }
    __syncthreads();
    pooled[tid] = s / msum_s;
    __syncthreads();
    if (tid < 10) {
        float acc = bfc[tid];
        for (int c = 0; c < 64; c++) acc += pooled[c] * Wfc[c * 10 + tid];
        lg[tid] = acc;
    }
    __syncthreads();
    if (tid == 0) {
        float mx = lg[0];
        for (int j = 1; j < 10; j++) mx = fmaxf(mx, lg[j]);
        float se = 0.f;
        for (int j = 0; j < 10; j++) se += __expf(lg[j] - mx);
        float lse = mx + __logf(se);
        for (int j = 0; j < 10; j++) out[b * 10 + j] = lg[j] - lse;
    }
}

// ---------------------------------------------------------------------------
extern "C" void kernel_launch(void* const* d_in, const int* in_sizes, int n_in,
                              void* d_out, int out_size, void* d_ws, size_t ws_size,
                              hipStream_t stream) {
    const float* inputs = (const float*)d_in[0];   // [2,1024,3]
    const float* Di     = (const float*)d_in[1];   // [2,8192,4096]
    const float* DiA    = (const float*)d_in[2];   // [2,4096,8192]
    const float* mask   = (const float*)d_in[3];   // [2,1024,1]
    const float* W_in   = (const float*)d_in[4];
    const float* b_in   = (const float*)d_in[5];
    const float* rn_W0  = (const float*)d_in[6];   // [5,64,64]
    const float* rn_b0  = (const float*)d_in[7];
    const float* rn_g0  = (const float*)d_in[8];
    const float* rn_be0 = (const float*)d_in[9];
    const float* rn_W1  = (const float*)d_in[10];
    const float* rn_b1  = (const float*)d_in[11];
    const float* rn_g1  = (const float*)d_in[12];
    const float* rn_be1 = (const float*)d_in[13];
    const float* bn2_g  = (const float*)d_in[14];
    const float* bn2_b  = (const float*)d_in[15];
    const float* W2     = (const float*)d_in[16];
    const float* b2     = (const float*)d_in[17];
    const float* Wfc    = (const float*)d_in[18];
    const float* bfc    = (const float*)d_in[19];
    float* out = (float*)d_out;

    // workspace layout (floats)
    float* ws    = (float*)d_ws;
    float* v     = ws;                // 2*1024*64 = 131072
    float* f     = v  + 131072;       // 2*2048*64 = 262144
    float* xt    = f  + 262144;       // 131072
    float* yt    = xt + 131072;       // 262144
    float* v2    = yt + 262144;       // 131072
    float* stats = v2 + 131072;       // 128

    projIn<<<512, 256, 0, stream>>>(inputs, W_in, b_in, v);
    zeroK<<<1024, 256, 0, stream>>>(f, 262144);

    for (int i = 0; i < 5; i++) {
        // vertex -> face
        bnStats<<<64, 256, 0, stream>>>(v, 2048, rn_g0 + i * 64, rn_be0 + i * 64, stats);
        transformK<<<64, 256, 0, stream>>>(v, stats, rn_W0 + i * 4096, rn_b0 + i * 64, xt);
        gemmN16<<<dim3(64, 2), 256, 0, stream>>>(Di, xt, f, 8192, 4096);   // f += Di@xt
        // face -> vertex
        bnStats<<<64, 256, 0, stream>>>(f, 4096, rn_g1 + i * 64, rn_be1 + i * 64, stats);
        transformK<<<128, 256, 0, stream>>>(f, stats, rn_W1 + i * 4096, rn_b1 + i * 64, yt);
        gemmN16<<<dim3(32, 2), 256, 0, stream>>>(DiA, yt, v, 4096, 8192);  // v += DiA@yt
    }

    bnStats<<<64, 256, 0, stream>>>(v, 2048, bn2_g, bn2_b, stats);
    transformK<<<64, 256, 0, stream>>>(v, stats, W2, b2, v2);
    headK<<<2, 64, 0, stream>>>(v2, mask, Wfc, bfc, out);
}